// SubGraph_32753420599416
// MI455X (gfx1250) — compile-verified
//
#include <hip/hip_runtime.h>
#include <math.h>

// ---------------------------------------------------------------------------
// CDNA5 (gfx1250) fused graph-encoder for the SubGraph reference.
//   layer: h = relu(x@w1+b1)@w2+b2 ; aggr = segment_max(h[src], dst) ;
//          x_next = [h | aggr]          (written directly in concat layout)
//   final: pooled = segment_max(x, cluster) ; out = pooled / (colnorm + 1e-6)
// GEMMs use V_WMMA_F32_16X16X4_F32 (native fp32 matrix path, wave32).
// Scatter-max uses monotone float->uint flip + global_atomic_max_u32.
// ---------------------------------------------------------------------------

typedef __attribute__((ext_vector_type(2))) float v2f;
typedef __attribute__((ext_vector_type(8))) float v8f;

__device__ __forceinline__ unsigned flipF(float f) {
    unsigned u = __float_as_uint(f);
    return (u >> 31) ? ~u : (u | 0x80000000u);
}
__device__ __forceinline__ float unflipF(unsigned u) {
    unsigned v = (u >> 31) ? (u & 0x7FFFFFFFu) : ~u;
    return __uint_as_float(v);
}
// flip(-inf) == 0x007FFFFF : initial value for max-scatter buffers
#define FLIP_NEG_INF 0x007FFFFFu

// ---------------------------------------------------------------------------
// Fused MLP: out[:, 0:C] = relu(x @ w1 + b1) @ w2 + b2   (out row stride 2C)
// One wave computes a 16-row tile. H = 64 hidden units fixed.
// ---------------------------------------------------------------------------
template <int C>
__global__ __launch_bounds__(64) void mlp_wmma_kernel(
    const float* __restrict__ x,    // [N, C] dense
    const float* __restrict__ w1,   // [C, 64]
    const float* __restrict__ b1,   // [64]
    const float* __restrict__ w2,   // [64, C]
    const float* __restrict__ b2,   // [C]
    float* __restrict__ out,        // [N, 2C] (left half written here)
    int nRows)
{
    constexpr int XS = C + 4;   // LDS row stride for X tile (keeps 16B align)
    constexpr int HS = 68;      // LDS row stride for h1 tile (bank-padded)
    __shared__ float ldsX[2][16 * XS];
    __shared__ float ldsH[2][16 * HS];

    const int wave = threadIdx.x >> 5;
    const int lane = threadIdx.x & 31;
    const int row0 = (blockIdx.x * 2 + wave) * 16;
    if (row0 >= nRows) return;          // uniform per wave: EXEC stays all-1s

    float* lx = ldsX[wave];
    float* lh = ldsH[wave];

    // ---- stage X tile (16 x C) into LDS, coalesced float4 loads ----
    {
        constexpr int QPR = C / 4;
        for (int idx = lane; idx < 16 * QPR; idx += 32) {
            int r = idx / QPR, q = idx - r * QPR;
            float4 v = *(const float4*)(x + (size_t)(row0 + r) * C + 4 * q);
            *(float4*)(lx + r * XS + 4 * q) = v;
        }
    }
    asm volatile("s_wait_dscnt 0" ::: "memory");

    const int half = lane >> 4;   // selects K pair {0,1} vs {2,3}
    const int l16  = lane & 15;

    // ---- GEMM1: h1[16x64] = relu(X @ W1 + b1) ----
    v8f acc[4];
#pragma unroll
    for (int t = 0; t < 4; ++t)
#pragma unroll
        for (int r = 0; r < 8; ++r) acc[t][r] = 0.0f;

    for (int kk = 0; kk < C; kk += 4) {
        v2f a = *(const v2f*)(lx + l16 * XS + kk + 2 * half);
        const int krow = kk + 2 * half;
#pragma unroll
        for (int t = 0; t < 4; ++t) {
            v2f b;
            b.x = w1[(size_t)krow * 64 + 16 * t + l16];
            b.y = w1[(size_t)(krow + 1) * 64 + 16 * t + l16];
            acc[t] = __builtin_amdgcn_wmma_f32_16x16x4_f32(
                false, a, false, b, (short)0, acc[t], false, false);
        }
    }

    // bias + relu, park h1 tile in LDS (row-major) for re-fragmenting as A
#pragma unroll
    for (int t = 0; t < 4; ++t) {
        float bias = b1[16 * t + l16];
#pragma unroll
        for (int r = 0; r < 8; ++r) {
            float v = acc[t][r] + bias;
            lh[(r + 8 * half) * HS + 16 * t + l16] = fmaxf(v, 0.0f);
        }
    }
    asm volatile("s_wait_dscnt 0" ::: "memory");

    // ---- GEMM2: out_tile[16xC] = h1 @ W2 + b2 ----
    for (int j = 0; j < C / 16; ++j) {
        v8f acc2;
#pragma unroll
        for (int r = 0; r < 8; ++r) acc2[r] = 0.0f;
#pragma unroll
        for (int kk = 0; kk < 64; kk += 4) {
            v2f a = *(const v2f*)(lh + l16 * HS + kk + 2 * half);
            const int krow = kk + 2 * half;
            v2f b;
            b.x = w2[(size_t)krow * C + 16 * j + l16];
            b.y = w2[(size_t)(krow + 1) * C + 16 * j + l16];
            acc2 = __builtin_amdgcn_wmma_f32_16x16x4_f32(
                false, a, false, b, (short)0, acc2, false, false);
        }
        float bias = b2[16 * j + l16];
#pragma unroll
        for (int r = 0; r < 8; ++r) {
            out[(size_t)(row0 + r + 8 * half) * (2 * C) + 16 * j + l16] =
                acc2[r] + bias;
        }
    }
}

// ---------------------------------------------------------------------------
// Fill a strided [rows x cols] region (stride, colOff in elements) with
// flip(-inf) so scatter-max can use u32 atomics.
// ---------------------------------------------------------------------------
__global__ void fill_flip_kernel(unsigned* __restrict__ base, long long rows,
                                 int cols, int stride, int colOff)
{
    long long gid = (long long)blockIdx.x * blockDim.x + threadIdx.x;
    long long total = rows * cols;
    if (gid >= total) return;
    long long r = gid / cols;
    int j = (int)(gid - r * cols);
    base[r * stride + colOff + j] = FLIP_NEG_INF;
}

// ---------------------------------------------------------------------------
// Edge scatter-max: for each edge e, channels 4q..4q+3:
//   aggrFlip[dst[e]] = max(aggrFlip[dst[e]], flip(h[src[e]]))
// h is the left half of xnext (stride 2C), aggr is the right half.
// ---------------------------------------------------------------------------
template <int C>
__global__ void scatter_max_kernel(float* xnext, const int* __restrict__ src,
                                   const int* __restrict__ dst, long long E)
{
    constexpr int QPR = C / 4;
    long long gid = (long long)blockIdx.x * blockDim.x + threadIdx.x;
    long long total = E * QPR;
    if (gid >= total) return;
    long long e = gid / QPR;
    int q = (int)(gid - e * QPR);
    int s = src[e], d = dst[e];
    float4 v = *(const float4*)(xnext + (size_t)s * (2 * C) + 4 * q);
    unsigned* ap = (unsigned*)(xnext + (size_t)d * (2 * C) + C + 4 * q);
    atomicMax(ap + 0, flipF(v.x));
    atomicMax(ap + 1, flipF(v.y));
    atomicMax(ap + 2, flipF(v.z));
    atomicMax(ap + 3, flipF(v.w));
}

// ---------------------------------------------------------------------------
// In-place unflip of a strided region: uint (flipped) -> float, with
// non-finite (no incoming edge => -inf) mapped to 0, per the reference.
// ---------------------------------------------------------------------------
__global__ void unflip_kernel(float* __restrict__ base, long long rows,
                              int cols, int stride, int colOff)
{
    long long gid = (long long)blockIdx.x * blockDim.x + threadIdx.x;
    long long total = rows * cols;
    if (gid >= total) return;
    long long r = gid / cols;
    int j = (int)(gid - r * cols);
    unsigned u = *(unsigned*)(base + r * stride + colOff + j);
    float f = unflipF(u);
    if (!isfinite(f)) f = 0.0f;
    base[r * stride + colOff + j] = f;
}

// ---------------------------------------------------------------------------
// Cluster max-pool: pooledFlip[cluster[n]][c] = max over nodes (512 channels)
// ---------------------------------------------------------------------------
__global__ void pool_max_kernel(const float* __restrict__ x,
                                const int* __restrict__ cluster,
                                unsigned* __restrict__ pooledFlip, int N)
{
    long long gid = (long long)blockIdx.x * blockDim.x + threadIdx.x;
    long long total = (long long)N * 128;   // 512 channels / 4
    if (gid >= total) return;
    int node = (int)(gid >> 7);
    int q = (int)(gid & 127);
    int cl = cluster[node];
    float4 v = *(const float4*)(x + (size_t)node * 512 + 4 * q);
    unsigned* ap = pooledFlip + (size_t)cl * 512 + 4 * q;
    atomicMax(ap + 0, flipF(v.x));
    atomicMax(ap + 1, flipF(v.y));
    atomicMax(ap + 2, flipF(v.z));
    atomicMax(ap + 3, flipF(v.w));
}

// Column L2 norms over the [NC, 512] pooled matrix (reads are coalesced:
// thread <-> column, rows walked together).
__global__ void colnorm_kernel(const float* __restrict__ pooled,
                               float* __restrict__ norms, int NC)
{
    int col = blockIdx.x * blockDim.x + threadIdx.x;
    if (col >= 512) return;
    float s = 0.0f;
    for (int r = 0; r < NC; ++r) {
        float v = pooled[(size_t)r * 512 + col];
        s += v * v;
    }
    norms[col] = sqrtf(s);
}

__global__ void scale_kernel(const float* __restrict__ pooled,
                             const float* __restrict__ norms,
                             float* __restrict__ out, long long total)
{
    long long gid = (long long)blockIdx.x * blockDim.x + threadIdx.x;
    if (gid >= total) return;
    int col = (int)(gid & 511);
    out[gid] = pooled[gid] / (norms[col] + 1e-6f);
}

// ---------------------------------------------------------------------------
static inline int gridFor(long long total, int block) {
    return (int)((total + block - 1) / block);
}

template <int C>
static void run_layer(const float* x, const float* w1, const float* b1,
                      const float* w2, const float* b2, float* xnext,
                      const int* src, const int* dst, int N, long long E,
                      hipStream_t stream)
{
    // fused MLP -> left half of xnext
    mlp_wmma_kernel<C><<<gridFor((long long)N, 32), 64, 0, stream>>>(
        x, w1, b1, w2, b2, xnext, N);
    // init right half to flip(-inf)
    fill_flip_kernel<<<gridFor((long long)N * C, 256), 256, 0, stream>>>(
        (unsigned*)xnext, N, C, 2 * C, C);
    // edge scatter-max
    scatter_max_kernel<C><<<gridFor(E * (C / 4), 256), 256, 0, stream>>>(
        xnext, src, dst, E);
    // unflip + zero-fill isolated nodes
    unflip_kernel<<<gridFor((long long)N * C, 256), 256, 0, stream>>>(
        xnext, N, C, 2 * C, C);
}

extern "C" void kernel_launch(void* const* d_in, const int* in_sizes, int n_in,
                              void* d_out, int out_size, void* d_ws, size_t ws_size,
                              hipStream_t stream)
{
    const float* x      = (const float*)d_in[0];
    const int*   eidx   = (const int*)d_in[1];     // [2, E]
    const int*   clus   = (const int*)d_in[2];
    const float* w1_0 = (const float*)d_in[3];
    const float* b1_0 = (const float*)d_in[4];
    const float* w2_0 = (const float*)d_in[5];
    const float* b2_0 = (const float*)d_in[6];
    const float* w1_1 = (const float*)d_in[7];
    const float* b1_1 = (const float*)d_in[8];
    const float* w2_1 = (const float*)d_in[9];
    const float* b2_1 = (const float*)d_in[10];
    const float* w1_2 = (const float*)d_in[11];
    const float* b1_2 = (const float*)d_in[12];
    const float* w2_2 = (const float*)d_in[13];
    const float* b2_2 = (const float*)d_in[14];

    const int N  = in_sizes[0] / 64;
    const long long E = (long long)in_sizes[1] / 2;
    const int NC = out_size / 512;
    const int* src = eidx;
    const int* dst = eidx + E;

    // workspace layout (floats)
    float* bufA   = (float*)d_ws;                    // N*512 (layer0 out N*128, layer2 out N*512)
    float* bufB   = bufA + (size_t)N * 512;          // N*256 (layer1 out)
    float* pooled = bufB + (size_t)N * 256;          // NC*512
    float* norms  = pooled + (size_t)NC * 512;       // 512

    // three encoder layers, channels 64 -> 128 -> 256 -> 512
    run_layer<64>(x, w1_0, b1_0, w2_0, b2_0, bufA, src, dst, N, E, stream);
    run_layer<128>(bufA, w1_1, b1_1, w2_1, b2_1, bufB, src, dst, N, E, stream);
    run_layer<256>(bufB, w1_2, b1_2, w2_2, b2_2, bufA, src, dst, N, E, stream);

    // cluster max-pool -> [NC, 512]
    fill_flip_kernel<<<gridFor((long long)NC * 512, 256), 256, 0, stream>>>(
        (unsigned*)pooled, NC, 512, 512, 0);
    pool_max_kernel<<<gridFor((long long)N * 128, 256), 256, 0, stream>>>(
        bufA, clus, (unsigned*)pooled, N);
    unflip_kernel<<<gridFor((long long)NC * 512, 256), 256, 0, stream>>>(
        pooled, NC, 512, 512, 0);

    // column-wise L2 normalization
    colnorm_kernel<<<2, 256, 0, stream>>>(pooled, norms, NC);
    scale_kernel<<<gridFor((long long)NC * 512, 256), 256, 0, stream>>>(
        pooled, norms, (float*)d_out, (long long)NC * 512);
}